// SACoFALoss_54898271978042
// MI455X (gfx1250) — compile-verified
//
#include <hip/hip_runtime.h>
#include <hip/hip_bf16.h>

typedef __attribute__((ext_vector_type(2))) float v2f;
typedef __attribute__((ext_vector_type(8))) float v8f;

#define A_DIM 1024
#define N_DIM 8192

// ws float layout: [0]=q0  [1]=q1  [2]=sum(w*nll)  [3]=sum(w)

__global__ void sacofa_init_ws(float* ws) {
    if (threadIdx.x < 4) ws[threadIdx.x] = 0.0f;
}

// q_l = d^T CV_l d, d = W[1,:] - W[0,:].
// 1024 waves: l(2) x jtile(64) x bsplit(8); 4 waves per 128-thread block.
// Each wave: V_WMMA_F32_16X16X4_F32 accumulating u[j0..j0+15] = sum_b d_b CV[b][j]
// over its b-range (A = d replicated over rows, B = coalesced 4x16 CV tile),
// then q += (1/16) * sum_j d_j * u_j  via wave shuffle + one f32 atomic.
__global__ void sacofa_quad_kernel(const float* __restrict__ W,
                                   const float* __restrict__ CV,
                                   float* __restrict__ ws) {
    __shared__ float sd[A_DIM];
    const int tid = threadIdx.x;
    for (int i = tid; i < A_DIM; i += blockDim.x)
        sd[i] = W[A_DIM + i] - W[i];
    __syncthreads();

    const int wid  = blockIdx.x * (blockDim.x >> 5) + (tid >> 5);
    const int lane = tid & 31;
    const int l    = wid >> 9;        // 0..1
    const int rem  = wid & 511;
    const int jt   = rem >> 3;        // 0..63
    const int bs   = rem & 7;         // 0..7
    const int j0   = jt * 16;
    const int col  = j0 + (lane & 15);
    const int khalf = (lane < 16) ? 0 : 2;
    const float* cv = CV + (size_t)l * A_DIM * A_DIM;

    v8f acc = {};
    const int b_beg = bs * 128;
    for (int b0 = b_beg; b0 < b_beg + 128; b0 += 4) {
        v2f a, b;
        // A (16x4 f32): lanes 0-15 -> K=0,1 ; lanes 16-31 -> K=2,3 (rows all equal d)
        a.x = sd[b0 + khalf];
        a.y = sd[b0 + khalf + 1];
        // B (4x16 f32): row-striped across lanes; contiguous 64B per half-wave
        b.x = cv[(size_t)(b0 + khalf) * A_DIM + col];
        b.y = cv[(size_t)(b0 + khalf + 1) * A_DIM + col];
        acc = __builtin_amdgcn_wmma_f32_16x16x4_f32(false, a, false, b,
                                                    (short)0, acc, false, false);
    }
    // Every D row equals u[col]; each u value appears 16x (8 VGPRs x 2 lane halves).
    float s = acc[0] + acc[1] + acc[2] + acc[3] + acc[4] + acc[5] + acc[6] + acc[7];
    float val = sd[col] * s;
    for (int off = 16; off > 0; off >>= 1)
        val += __shfl_down(val, off, 32);
    if (lane == 0)
        atomicAdd(&ws[l], val * (1.0f / 16.0f));
}

// Weighted 2-class cross-entropy using precomputed q0/q1.
__global__ void sacofa_loss_kernel(const float* __restrict__ y,
                                   const int* __restrict__ tgt,
                                   const float* __restrict__ ratio,
                                   const float* __restrict__ ws) {
    __shared__ float snum[256];
    __shared__ float sden[256];
    float* acc = (float*)ws;  // acc[2], acc[3]
    const float hr = 0.5f * ratio[0];
    const float q0 = ws[0], q1 = ws[1];

    float num = 0.0f, den = 0.0f;
    for (int n = blockIdx.x * blockDim.x + threadIdx.x; n < N_DIM;
         n += gridDim.x * blockDim.x) {
        const int t = tgt[n];
        float a0 = y[2 * n + 0];
        float a1 = y[2 * n + 1];
        // target=0: sigma2=(0,q0); target=1: sigma2=(q1,0)
        if (t == 0) a1 += hr * q0; else a0 += hr * q1;
        float m   = fmaxf(a0, a1);
        float lse = m + logf(expf(a0 - m) + expf(a1 - m));
        float at  = (t == 0) ? a0 : a1;
        float w   = (t == 0) ? 1.0f : 0.5f;   // CE_CLASS_WEIGHTS
        num += w * (lse - at);
        den += w;
    }
    snum[threadIdx.x] = num;
    sden[threadIdx.x] = den;
    __syncthreads();
    for (int s = blockDim.x >> 1; s > 0; s >>= 1) {
        if ((int)threadIdx.x < s) {
            snum[threadIdx.x] += snum[threadIdx.x + s];
            sden[threadIdx.x] += sden[threadIdx.x + s];
        }
        __syncthreads();
    }
    if (threadIdx.x == 0) {
        atomicAdd(&acc[2], snum[0]);
        atomicAdd(&acc[3], sden[0]);
    }
}

__global__ void sacofa_finalize(const float* __restrict__ ws,
                                float* __restrict__ out) {
    if (threadIdx.x == 0 && blockIdx.x == 0)
        out[0] = ws[2] / ws[3];
}

extern "C" void kernel_launch(void* const* d_in, const int* in_sizes, int n_in,
                              void* d_out, int out_size, void* d_ws, size_t ws_size,
                              hipStream_t stream) {
    (void)in_sizes; (void)n_in; (void)out_size; (void)ws_size;
    const float* fc_weight = (const float*)d_in[0];
    // d_in[1] = features: unused by the reference computation
    const float* y         = (const float*)d_in[2];
    const int*   target_x  = (const int*)d_in[3];
    const float* CoVar     = (const float*)d_in[4];
    const float* ratio     = (const float*)d_in[5];
    float* out = (float*)d_out;
    float* ws  = (float*)d_ws;

    sacofa_init_ws<<<1, 32, 0, stream>>>(ws);
    sacofa_quad_kernel<<<256, 128, 0, stream>>>(fc_weight, CoVar, ws);
    sacofa_loss_kernel<<<32, 256, 0, stream>>>(y, target_x, ratio, ws);
    sacofa_finalize<<<1, 1, 0, stream>>>(ws, out);
}